// SWAStatefulChunk1Single_90194313216383
// MI455X (gfx1250) — compile-verified
//
#include <hip/hip_runtime.h>
#include <hip/hip_bf16.h>
#include <math.h>

#define HID   2048
#define NH    8
#define HKV   2
#define HD    256
#define NREP  4
#define WWIN  512
#define CTX   2048
#define IMED  4096
#define NL    8
#define LTOT  30
#define PLD   256
#define FEPS  1e-6f
#define PLI_SCALE 0.7071067811865476f
#define PLE_SCALE 0.022097086912079608f

typedef __attribute__((ext_vector_type(2))) float v2f;
typedef __attribute__((ext_vector_type(8))) float v8f;

// -------------------- block reduction helpers (wave32) --------------------
__device__ __forceinline__ float block_allreduce_sum(float v) {
    __shared__ float red[32];
    const int lane = threadIdx.x & 31;
    const int wid  = threadIdx.x >> 5;
    #pragma unroll
    for (int o = 16; o > 0; o >>= 1) v += __shfl_xor(v, o, 32);
    if (lane == 0) red[wid] = v;
    __syncthreads();
    const int nw = (blockDim.x + 31) >> 5;
    if (threadIdx.x == 0) {
        float s = 0.f;
        for (int i = 0; i < nw; ++i) s += red[i];
        red[0] = s;
    }
    __syncthreads();
    const float r = red[0];
    __syncthreads();
    return r;
}

__device__ __forceinline__ float block_allreduce_max(float v) {
    __shared__ float redm[32];
    const int lane = threadIdx.x & 31;
    const int wid  = threadIdx.x >> 5;
    #pragma unroll
    for (int o = 16; o > 0; o >>= 1) v = fmaxf(v, __shfl_xor(v, o, 32));
    if (lane == 0) redm[wid] = v;
    __syncthreads();
    const int nw = (blockDim.x + 31) >> 5;
    if (threadIdx.x == 0) {
        float s = -3.0e38f;
        for (int i = 0; i < nw; ++i) s = fmaxf(s, redm[i]);
        redm[0] = s;
    }
    __syncthreads();
    const float r = redm[0];
    __syncthreads();
    return r;
}

__device__ __forceinline__ float gelu_tanh(float x) {
    const float c = 0.7978845608028654f; // sqrt(2/pi)
    return 0.5f * x * (1.f + tanhf(c * (x + 0.044715f * x * x * x)));
}

// async copy 16B global -> LDS (per active lane), tracked by ASYNCcnt
__device__ __forceinline__ void async_ldsb128(unsigned lds_off, const float* g) {
    asm volatile("global_load_async_to_lds_b128 %0, %1, off"
                 :: "v"(lds_off), "v"(g) : "memory");
}

// -------------------- WMMA GEMV: y[N] = W[N,K] @ x[K] --------------------
// grid.x = N/64, block = 128 (4 waves). Wave w owns rows [blk*64 + w*16, +16).
// Double-buffered LDS tiles (64 rows x 32 cols, padded stride 36) filled with
// GLOBAL_LOAD_ASYNC_TO_LDS_B128 (no VGPR round trip); exactly 5 async
// instructions per wave per stage -> s_wait_asynccnt 0x5 lets the next stage's
// HBM traffic overlap the current tile's V_WMMA_F32_16X16X4_F32 work.
// WMMA A (16x4 fp32): lanes 0-15 hold {K=k,k+1} of row m=lane, lanes 16-31 hold
// {K=k+2,k+3}. B (4x16) is x broadcast across the 16 N columns, so every C
// column holds the GEMV result; lanes 0 and 16 write the 16 row results.
#define GBK  32
#define GLDP 36
__global__ void gemv_wmma_kernel(const float* __restrict__ W,
                                 const float* __restrict__ x,
                                 float* __restrict__ y, int K) {
    __shared__ float sW[2][64 * GLDP];
    __shared__ float sX[2][GBK];
    const int t    = threadIdx.x;
    const int lane = t & 31;
    const int wave = t >> 5;
    const int rowBlock = blockIdx.x * 64;
    const int m     = lane & 15;
    const int khalf = (lane >> 4) << 1; // 0 or 2

    // one stage: 64x32 W tile (4 b128/thread = 4 insts/wave) + 32-float x chunk
    // (1 predicated inst/wave) -> 5 ASYNCcnt increments per wave
    auto stage = [&](int k0, int buf) {
        #pragma unroll
        for (int j = 0; j < 4; ++j) {
            const int i  = t + j * 128;       // 0..511
            const int r  = i >> 3;            // row 0..63
            const int c4 = i & 7;             // float4 col 0..7
            const unsigned lds = (unsigned)(size_t)&sW[buf][r * GLDP + (c4 << 2)];
            const float* g = W + (size_t)(rowBlock + r) * K + k0 + (c4 << 2);
            async_ldsb128(lds, g);
        }
        if (lane < 2) {
            const int idx = (wave << 1) | lane;   // 0..7
            const unsigned lds = (unsigned)(size_t)&sX[buf][idx << 2];
            const float* g = x + k0 + (idx << 2);
            async_ldsb128(lds, g);
        }
    };

    v8f acc = {};
    int buf = 0;
    stage(0, 0);
    for (int k0 = 0; k0 < K; k0 += GBK) {
        if (k0 + GBK < K) {
            stage(k0 + GBK, buf ^ 1);
            asm volatile("s_wait_asynccnt 0x5" ::: "memory"); // prev stage done
        } else {
            asm volatile("s_wait_asynccnt 0x0" ::: "memory");
        }
        __syncthreads();
        const float* wrow = &sW[buf][(wave * 16 + m) * GLDP];
        const float* xb   = sX[buf];
        #pragma unroll
        for (int kk = 0; kk < GBK; kk += 4) {
            v2f a, b;
            a.x = wrow[kk + khalf];
            a.y = wrow[kk + khalf + 1];
            b.x = xb[kk + khalf];
            b.y = xb[kk + khalf + 1];
            acc = __builtin_amdgcn_wmma_f32_16x16x4_f32(
                false, a, false, b, (short)0, acc, false, false);
        }
        __syncthreads(); // protect buf before it is refilled
        buf ^= 1;
    }
    // C layout: VGPR j, lanes 0-15 -> M=j (N=lane), lanes 16-31 -> M=j+8.
    if (lane == 0 || lane == 16) {
        const int rbase = rowBlock + wave * 16 + ((lane >> 4) << 3);
        #pragma unroll
        for (int j = 0; j < 8; ++j) y[rbase + j] = acc[j];
    }
}

// -------------------- RMSNorm kernels (n = 2048, 1 block x 256) ----------
__global__ void rmsnorm_w_kernel(const float* __restrict__ in,
                                 const float* __restrict__ w,
                                 float* __restrict__ out) {
    const int t = threadIdx.x;
    float v[8]; float ss = 0.f;
    #pragma unroll
    for (int j = 0; j < 8; ++j) { v[j] = in[t + j * 256]; ss += v[j] * v[j]; }
    ss = block_allreduce_sum(ss);
    const float rn = rsqrtf(ss * (1.f / HID) + FEPS);
    #pragma unroll
    for (int j = 0; j < 8; ++j) {
        const int idx = t + j * 256;
        out[idx] = v[j] * rn * (1.f + w[idx]);
    }
}

// hs = (hs + rms(g)*(1+w)) * (ls ? *ls : 1)
__global__ void rms_add_kernel(float* __restrict__ hs,
                               const float* __restrict__ g,
                               const float* __restrict__ w,
                               const float* __restrict__ ls) {
    const int t = threadIdx.x;
    float v[8]; float ss = 0.f;
    #pragma unroll
    for (int j = 0; j < 8; ++j) { v[j] = g[t + j * 256]; ss += v[j] * v[j]; }
    ss = block_allreduce_sum(ss);
    const float rn = rsqrtf(ss * (1.f / HID) + FEPS);
    const float s  = ls ? ls[0] : 1.f;
    #pragma unroll
    for (int j = 0; j < 8; ++j) {
        const int idx = t + j * 256;
        hs[idx] = (hs[idx] + v[j] * rn * (1.f + w[idx])) * s;
    }
}

// -------------------- q/k/v head norm + rope + cache write ----------------
// blocks 0-7: q heads (in-place), 8-9: k heads -> cacheK@pos, 10-11: v -> cacheV@pos
__global__ void qkv_rope_kernel(float* __restrict__ q,
                                const float* __restrict__ kin,
                                const float* __restrict__ vin,
                                const float* __restrict__ qw,
                                const float* __restrict__ kw,
                                const float* __restrict__ cs,
                                const float* __restrict__ sn,
                                float* __restrict__ cacheK,
                                float* __restrict__ cacheV,
                                const int* __restrict__ posp) {
    __shared__ float xs[HD];
    const int b = blockIdx.x;
    const int d = threadIdx.x;
    const int pos = posp[0];
    float x;
    if (b < 8)       x = q[b * HD + d];
    else if (b < 10) x = kin[(b - 8) * HD + d];
    else             x = vin[(b - 10) * HD + d];
    xs[d] = x;
    const float ss = block_allreduce_sum(x * x); // barrier also publishes xs
    const float rn = rsqrtf(ss * (1.f / HD) + FEPS);
    if (b < 10) {
        const float* w = (b < 8) ? qw : kw;
        const float nv  = x * rn * (1.f + w[d]);
        const int   d2  = (d < 128) ? d + 128 : d - 128;
        const float pv  = xs[d2] * rn * (1.f + w[d2]);
        const float sgn = (d < 128) ? -1.f : 1.f;
        const float o   = nv * cs[d] + sgn * pv * sn[d];
        if (b < 8) q[b * HD + d] = o;
        else cacheK[(size_t)(b - 8) * CTX * HD + (size_t)pos * HD + d] = o;
    } else {
        cacheV[(size_t)(b - 10) * CTX * HD + (size_t)pos * HD + d] = x * rn;
    }
}

// -------------------- attention -------------------------------------------
// one wave per score; grid = NH * S/8, block 256 (8 waves); b128 loads
__global__ void attn_scores_kernel(const float* __restrict__ q,
                                   const float* __restrict__ Kbase,
                                   const float* __restrict__ mask,
                                   float* __restrict__ sc, int S) {
    const int wave = threadIdx.x >> 5, lane = threadIdx.x & 31;
    const int per  = S >> 3;
    const int h    = blockIdx.x / per;
    const int s    = (blockIdx.x % per) * 8 + wave;
    const int kv   = h >> 2; // NREP = 4
    const float4* qh   = reinterpret_cast<const float4*>(q + h * HD) + lane * 2;
    const float4* krow = reinterpret_cast<const float4*>(
        Kbase + (size_t)kv * CTX * HD + (size_t)s * HD) + lane * 2;
    float acc = 0.f;
    #pragma unroll
    for (int j = 0; j < 2; ++j) {
        const float4 a = qh[j];
        const float4 b = krow[j];
        acc += a.x * b.x + a.y * b.y + a.z * b.z + a.w * b.w;
    }
    #pragma unroll
    for (int o = 16; o > 0; o >>= 1) acc += __shfl_xor(acc, o, 32);
    if (lane == 0) sc[(size_t)h * S + s] = acc + mask[s];
}

__global__ void softmax_kernel(float* __restrict__ sc, int S) {
    float* p = sc + (size_t)blockIdx.x * S;
    float mx = -3.0e38f;
    for (int i = threadIdx.x; i < S; i += 256) mx = fmaxf(mx, p[i]);
    mx = block_allreduce_max(mx);
    float sum = 0.f;
    for (int i = threadIdx.x; i < S; i += 256) {
        const float e = __expf(p[i] - mx);
        p[i] = e; sum += e;
    }
    sum = block_allreduce_sum(sum);
    const float inv = 1.f / sum;
    for (int i = threadIdx.x; i < S; i += 256) p[i] *= inv;
}

// out[h*HD+d] = sum_s probs[h][s] * V[kv(h)][s][d] ; grid NH, block HD
__global__ void attn_v_kernel(const float* __restrict__ probs,
                              const float* __restrict__ Vbase,
                              float* __restrict__ out, int S) {
    const int h = blockIdx.x, d = threadIdx.x;
    const int kv = h >> 2;
    const float* V = Vbase + (size_t)kv * CTX * HD;
    const float* p = probs + (size_t)h * S;
    float acc = 0.f;
    for (int s = 0; s < S; ++s) {
        if ((s & 15) == 0 && s + 16 < S)
            __builtin_prefetch(&V[(size_t)(s + 16) * HD + d], 0, 0);
        acc += p[s] * V[(size_t)s * HD + d];
    }
    out[h * HD + d] = acc;
}

// -------------------- elementwise ------------------------------------------
__global__ void gelu_mul_kernel(const float* __restrict__ g,
                                const float* __restrict__ u,
                                float* __restrict__ o, int n) {
    const int i = blockIdx.x * blockDim.x + threadIdx.x;
    if (i < n) o[i] = gelu_tanh(g[i]) * u[i];
}

// PLE: per-group rms over PLD, * (1+nw), + raw, * PLI_SCALE; grid LTOT, block PLD
__global__ void ple_norm_kernel(const float* __restrict__ proj,
                                const float* __restrict__ nw,
                                const float* __restrict__ raw,
                                float* __restrict__ plc) {
    const int g = blockIdx.x, d = threadIdx.x;
    const float x = proj[g * PLD + d] * PLE_SCALE;
    const float ss = block_allreduce_sum(x * x);
    const float rn = rsqrtf(ss * (1.f / PLD) + FEPS);
    plc[g * PLD + d] = (x * rn * (1.f + nw[d]) + raw[g * PLD + d]) * PLI_SCALE;
}

// ===========================================================================
extern "C" void kernel_launch(void* const* d_in, const int* in_sizes, int n_in,
                              void* d_out, int out_size, void* d_ws, size_t ws_size,
                              hipStream_t stream) {
    (void)in_sizes; (void)n_in; (void)out_size; (void)ws_size;
    const float* hidden       = (const float*)d_in[0];
    const float* plraw        = (const float*)d_in[1];
    const float* cos_s        = (const float*)d_in[2];
    const float* sin_s        = (const float*)d_in[3];
    const float* cos_f        = (const float*)d_in[4];
    const float* sin_f        = (const float*)d_in[5];
    const float* mask_f       = (const float*)d_in[6];
    const float* mask_s       = (const float*)d_in[7];
    const float* cache_in     = (const float*)d_in[8];
    const float* st13k        = (const float*)d_in[9];
    const float* st13v        = (const float*)d_in[10];
    const float* st14k        = (const float*)d_in[11];
    const float* st14v        = (const float*)d_in[12];
    const float* in_ln        = (const float*)d_in[13];
    const float* post_attn_ln = (const float*)d_in[14];
    const float* pre_ff_ln    = (const float*)d_in[15];
    const float* post_ff_ln   = (const float*)d_in[16];
    const float* post_pl_ln   = (const float*)d_in[17];
    const float* q_norm_w     = (const float*)d_in[18];
    const float* k_norm_w     = (const float*)d_in[19];
    const float* q_w          = (const float*)d_in[20];
    const float* k_w          = (const float*)d_in[21];
    const float* v_w          = (const float*)d_in[22];
    const float* o_w          = (const float*)d_in[23];
    const float* gate_w       = (const float*)d_in[24];
    const float* up_w         = (const float*)d_in[25];
    const float* down_w       = (const float*)d_in[26];
    const float* pl_gate_w    = (const float*)d_in[27];
    const float* pl_proj_w    = (const float*)d_in[28];
    const float* layer_scalar = (const float*)d_in[29];
    const float* ple_w        = (const float*)d_in[30];
    const float* ple_norm_w   = (const float*)d_in[31];
    const int*   cur_pos      = (const int*)d_in[32];
    const int*   ring_pos     = (const int*)d_in[33];

    float* out = (float*)d_out;
    float* outCache = out + 2048;
    const size_t CACHE_N = (size_t)2 * NL * HKV * CTX * HD;  // 16,777,216
    const size_t ST13 = (size_t)HKV * WWIN * HD;             // 262,144
    const size_t ST14 = (size_t)HKV * CTX * HD;              // 1,048,576
    float* o13k = outCache + CACHE_N;
    float* o13v = o13k + ST13;
    float* o14k = o13v + ST13;
    float* o14v = o14k + ST14;

    // pass-through outputs (graph-capture safe async d2d copies)
    hipMemcpyAsync(outCache, cache_in, CACHE_N * sizeof(float), hipMemcpyDeviceToDevice, stream);
    hipMemcpyAsync(o13k, st13k, ST13 * sizeof(float), hipMemcpyDeviceToDevice, stream);
    hipMemcpyAsync(o13v, st13v, ST13 * sizeof(float), hipMemcpyDeviceToDevice, stream);
    hipMemcpyAsync(o14k, st14k, ST14 * sizeof(float), hipMemcpyDeviceToDevice, stream);
    hipMemcpyAsync(o14v, st14v, ST14 * sizeof(float), hipMemcpyDeviceToDevice, stream);

    // workspace layout (floats)
    float* ws    = (float*)d_ws;
    float* hs    = ws;            // 2048
    float* hn    = ws + 2048;     // 2048
    float* qb    = ws + 4096;     // 2048
    float* kb    = ws + 6144;     // 512
    float* vb    = ws + 6656;     // 512
    float* sc    = ws + 7168;     // 16384
    float* attn  = ws + 23552;    // 2048
    float* g1    = ws + 25600;    // 4096
    float* g2    = ws + 29696;    // 4096
    float* g3    = ws + 33792;    // 4096
    float* plc   = ws + 37888;    // 7680
    float* pproj = ws + 45568;    // 7680
    float* tmp   = ws + 53248;    // 2048
    float* plg   = ws + 55296;    // 256
    float* plg2  = ws + 55552;    // 256

    hipMemcpyAsync(hs, hidden, HID * sizeof(float), hipMemcpyDeviceToDevice, stream);

    // ---- per-layer embedding path ----
    gemv_wmma_kernel<<<(LTOT * PLD) / 64, 128, 0, stream>>>(ple_w, hs, pproj, HID);
    ple_norm_kernel<<<LTOT, PLD, 0, stream>>>(pproj, ple_norm_w, plraw, plc);

    for (int i = 0; i < NL; ++i) {
        const bool full = ((i + 1) % 6) == 0;
        const int  S    = full ? CTX : WWIN;
        const float* mask = full ? mask_f : mask_s;
        const float* cs   = full ? cos_f : cos_s;
        const float* sn   = full ? sin_f : sin_s;
        const int*   posp = full ? cur_pos : ring_pos;
        float* cK = outCache + (size_t)(2 * i) * HKV * CTX * HD;
        float* cV = outCache + (size_t)(2 * i + 1) * HKV * CTX * HD;

        // attention block
        rmsnorm_w_kernel<<<1, 256, 0, stream>>>(hs, in_ln + (size_t)i * HID, hn);
        gemv_wmma_kernel<<<(NH * HD) / 64, 128, 0, stream>>>(q_w + (size_t)i * NH * HD * HID, hn, qb, HID);
        gemv_wmma_kernel<<<(HKV * HD) / 64, 128, 0, stream>>>(k_w + (size_t)i * HKV * HD * HID, hn, kb, HID);
        gemv_wmma_kernel<<<(HKV * HD) / 64, 128, 0, stream>>>(v_w + (size_t)i * HKV * HD * HID, hn, vb, HID);
        qkv_rope_kernel<<<12, HD, 0, stream>>>(qb, kb, vb,
                                               q_norm_w + (size_t)i * HD, k_norm_w + (size_t)i * HD,
                                               cs, sn, cK, cV, posp);
        attn_scores_kernel<<<NH * (S / 8), 256, 0, stream>>>(qb, cK, mask, sc, S);
        softmax_kernel<<<NH, 256, 0, stream>>>(sc, S);
        attn_v_kernel<<<NH, HD, 0, stream>>>(sc, cV, attn, S);
        gemv_wmma_kernel<<<HID / 64, 128, 0, stream>>>(o_w + (size_t)i * HID * NH * HD, attn, tmp, NH * HD);
        rms_add_kernel<<<1, 256, 0, stream>>>(hs, tmp, post_attn_ln + (size_t)i * HID, nullptr);

        // MLP block (GEGLU)
        rmsnorm_w_kernel<<<1, 256, 0, stream>>>(hs, pre_ff_ln + (size_t)i * HID, hn);
        gemv_wmma_kernel<<<IMED / 64, 128, 0, stream>>>(gate_w + (size_t)i * IMED * HID, hn, g1, HID);
        gemv_wmma_kernel<<<IMED / 64, 128, 0, stream>>>(up_w + (size_t)i * IMED * HID, hn, g2, HID);
        gelu_mul_kernel<<<IMED / 256, 256, 0, stream>>>(g1, g2, g3, IMED);
        gemv_wmma_kernel<<<HID / 64, 128, 0, stream>>>(down_w + (size_t)i * HID * IMED, g3, tmp, IMED);
        rms_add_kernel<<<1, 256, 0, stream>>>(hs, tmp, post_ff_ln + (size_t)i * HID, nullptr);

        // per-layer gate/proj block (+ layer_scalar)
        gemv_wmma_kernel<<<PLD / 64, 128, 0, stream>>>(pl_gate_w + (size_t)i * PLD * HID, hs, plg, HID);
        gelu_mul_kernel<<<1, 256, 0, stream>>>(plg, plc + (size_t)i * PLD, plg2, PLD);
        gemv_wmma_kernel<<<HID / 64, 128, 0, stream>>>(pl_proj_w + (size_t)i * HID * PLD, plg2, tmp, PLD);
        rms_add_kernel<<<1, 256, 0, stream>>>(hs, tmp, post_pl_ln + (size_t)i * HID, layer_scalar + i);
    }

    hipMemcpyAsync(out, hs, HID * sizeof(float), hipMemcpyDeviceToDevice, stream);
}